// GATDP_78185584657133
// MI455X (gfx1250) — compile-verified
//
#include <hip/hip_runtime.h>
#include <cmath>

// ---------------------------------------------------------------------------
// GAT (2 layers) + DP noise for MI455X (gfx1250, wave32).
//  - Dense node transforms via V_WMMA_F32_16X16X4_F32, 32x64 tile per wave,
//    pre-paired B operands (one b64 load per B fragment).
//  - Edge softmax via 3-pass (atomic max, exp+denom, scatter) — atomics land
//    in the 192MB L2 (accumulators are <=102MB).
// ---------------------------------------------------------------------------

typedef __attribute__((ext_vector_type(2))) float v2f;
typedef __attribute__((ext_vector_type(8))) float v8f;

#define WMMA_F32(a, b, c) \
    __builtin_amdgcn_wmma_f32_16x16x4_f32(false, (a), false, (b), (short)0, (c), false, false)

// Pre-pair B rows: Bp[p][n] = {B[2p][n], B[2p+1][n]}  (p = 0..K/2-1)
__global__ void repack_B_kernel(const float* __restrict__ B, float* __restrict__ Bp,
                                int K, int Ncols) {
    int i = blockIdx.x * blockDim.x + threadIdx.x;
    int total = (K >> 1) * Ncols;
    if (i >= total) return;
    int p = i / Ncols, n = i % Ncols;
    v2f b;
    b.x = B[(size_t)(2 * p) * Ncols + n];
    b.y = B[(size_t)(2 * p + 1) * Ncols + n];
    ((v2f*)Bp)[i] = b;
}

// ---------------- WMMA GEMM: C[M,Ncols] = A[M,K] * B[K,Ncols] --------------
// B passed pre-paired (repack_B_kernel). One wave -> 32(M) x 64(N) tile.
// Requires M%32==0, Ncols%64==0, K%4==0.
__global__ void __launch_bounds__(256)
gemm_wmma_f32(const float* __restrict__ A, const float* __restrict__ Bp,
              float* __restrict__ C, int M, int K, int Ncols) {
    const int wave   = blockIdx.x * (blockDim.x >> 5) + (threadIdx.x >> 5);
    const int tilesN = Ncols >> 6;
    const int nTiles = (M >> 5) * tilesN;
    if (wave >= nTiles) return;                // wave-uniform exit: EXEC stays full
    const int m0 = (wave / tilesN) << 5;
    const int n0 = (wave % tilesN) << 6;
    const int lane = threadIdx.x & 31;
    const int lm   = lane & 15;                // M (A) / N (B) index within fragment
    const int hi   = lane >> 4;                // which K pair of the k-step

    const float* arow0 = A + (size_t)(m0 + lm) * K + 2 * hi;       // rows m0..m0+15
    const float* arow1 = arow0 + (size_t)16 * K;                   // rows m0+16..m0+31
    const v2f*   bbase = (const v2f*)Bp + (size_t)hi * Ncols + n0 + lm;

    v8f c00 = {}, c01 = {}, c02 = {}, c03 = {};
    v8f c10 = {}, c11 = {}, c12 = {}, c13 = {};

#pragma unroll 2
    for (int k0 = 0; k0 < K; k0 += 4) {
        v2f a0 = *(const v2f*)(arow0 + k0);                        // A[m][k0+2hi, +1]
        v2f a1 = *(const v2f*)(arow1 + k0);
        const v2f* bp = bbase + (size_t)(k0 >> 1) * Ncols;
        v2f b0 = bp[0];
        v2f b1 = bp[16];
        v2f b2 = bp[32];
        v2f b3 = bp[48];
        c00 = WMMA_F32(a0, b0, c00);
        c10 = WMMA_F32(a1, b0, c10);
        c01 = WMMA_F32(a0, b1, c01);
        c11 = WMMA_F32(a1, b1, c11);
        c02 = WMMA_F32(a0, b2, c02);
        c12 = WMMA_F32(a1, b2, c12);
        c03 = WMMA_F32(a0, b3, c03);
        c13 = WMMA_F32(a1, b3, c13);
    }

    // C/D layout: vgpr r -> row (r + 8*hi), col lm
    float* crow0 = C + (size_t)(m0 + 8 * hi) * Ncols + n0 + lm;
    float* crow1 = crow0 + (size_t)16 * Ncols;
    v8f ca[4] = {c00, c01, c02, c03};
    v8f cb[4] = {c10, c11, c12, c13};
#pragma unroll
    for (int t = 0; t < 4; ++t)
#pragma unroll
        for (int r = 0; r < 8; ++r) {
            crow0[(size_t)r * Ncols + t * 16] = ca[t][r];
            crow1[(size_t)r * Ncols + t * 16] = cb[t][r];
        }
}

// ---------------- helpers ---------------------------------------------------
__global__ void fill_kernel(float* __restrict__ p, long total, float v) {
    long i = (long)blockIdx.x * blockDim.x + threadIdx.x;
    if (i < total) p[i] = v;
}

__global__ void bias_init_kernel(float* __restrict__ out, const float* __restrict__ b,
                                 long total, int C) {
    long i = (long)blockIdx.x * blockDim.x + threadIdx.x;
    if (i < total) out[i] = b[i % C];
}

// el/er per (node, head): dot(h[n,hd,:], al[hd,:]) and dot(., ar[hd,:])
__global__ void attn_scores_kernel(const float* __restrict__ h,
                                   const float* __restrict__ al,
                                   const float* __restrict__ ar,
                                   float* __restrict__ el, float* __restrict__ er,
                                   int N, int H, int D) {
    long t = (long)blockIdx.x * blockDim.x + threadIdx.x;
    if (t >= (long)N * H) return;
    int n = (int)(t / H), hd = (int)(t % H);
    const float4* hp  = (const float4*)(h + (size_t)n * H * D + (size_t)hd * D);
    const float4* alp = (const float4*)(al + (size_t)hd * D);
    const float4* arp = (const float4*)(ar + (size_t)hd * D);
    float sl = 0.f, sr = 0.f;
    for (int i = 0; i < (D >> 2); ++i) {
        float4 hv = hp[i], a = alp[i], r = arp[i];
        sl += hv.x * a.x + hv.y * a.y + hv.z * a.z + hv.w * a.w;
        sr += hv.x * r.x + hv.y * r.y + hv.z * r.z + hv.w * r.w;
    }
    el[t] = sl; er[t] = sr;
}

__device__ inline void atomicMaxFloat(float* addr, float v) {
    // IEEE ordering trick; valid with -inf init and mixed signs.
    if (v >= 0.f) atomicMax((int*)addr, __float_as_int(v));
    else          atomicMin((unsigned int*)addr, __float_as_uint(v));
}

__global__ void edge_max_kernel(const int* __restrict__ src, const int* __restrict__ dst,
                                const float* __restrict__ el, const float* __restrict__ er,
                                float* __restrict__ emax, long EH, int H) {
    long t = (long)blockIdx.x * blockDim.x + threadIdx.x;
    if (t >= EH) return;
    long e = t / H; int hd = (int)(t % H);
    int s = src[e], d = dst[e];
    float v = el[(size_t)s * H + hd] + er[(size_t)d * H + hd];
    v = (v > 0.f) ? v : 0.2f * v;                 // leaky_relu(., 0.2)
    atomicMaxFloat(&emax[(size_t)d * H + hd], v);
}

__global__ void edge_exp_kernel(const int* __restrict__ src, const int* __restrict__ dst,
                                const float* __restrict__ el, const float* __restrict__ er,
                                const float* __restrict__ emax, float* __restrict__ denom,
                                float* __restrict__ abuf, long EH, int H) {
    long t = (long)blockIdx.x * blockDim.x + threadIdx.x;
    if (t >= EH) return;
    long e = t / H; int hd = (int)(t % H);
    int s = src[e], d = dst[e];
    float v = el[(size_t)s * H + hd] + er[(size_t)d * H + hd];
    v = (v > 0.f) ? v : 0.2f * v;
    float a = __expf(v - emax[(size_t)d * H + hd]);
    abuf[t] = a;
    atomicAdd(&denom[(size_t)d * H + hd], a);
}

// scatter: out[dst] += alpha * h[src]; one thread per float4 chunk of a message
__global__ void edge_scatter_kernel(const int* __restrict__ src, const int* __restrict__ dst,
                                    const float* __restrict__ abuf, const float* __restrict__ denom,
                                    const float* __restrict__ h, float* __restrict__ out,
                                    long E, int H, int D) {
    const int chunks = (H * D) >> 2;
    long t = (long)blockIdx.x * blockDim.x + threadIdx.x;
    if (t >= E * (long)chunks) return;
    long e = t / chunks; int r = (int)(t % chunks);
    int hd = (r << 2) / D;
    int s = src[e], d = dst[e];
    float alpha = abuf[e * H + hd] / denom[(size_t)d * H + hd];
    float4 hv = *(const float4*)(h + (size_t)s * H * D + ((size_t)r << 2));
    float* op = out + (size_t)d * H * D + ((size_t)r << 2);
    atomicAdd(op + 0, hv.x * alpha);
    atomicAdd(op + 1, hv.y * alpha);
    atomicAdd(op + 2, hv.z * alpha);
    atomicAdd(op + 3, hv.w * alpha);
}

// h1 = leaky_relu(acc0 + b0, 0.01) + scale*noise  (in place)
__global__ void h1_kernel(float* __restrict__ h1, const float* __restrict__ b0,
                          const float* __restrict__ noise, float scale, long total, int C) {
    long i = (long)blockIdx.x * blockDim.x + threadIdx.x;
    if (i >= total) return;
    float v = h1[i] + b0[i % C];
    v = (v > 0.f) ? v : 0.01f * v;
    h1[i] = v + scale * noise[i];
}

// ---------------------------------------------------------------------------
extern "C" void kernel_launch(void* const* d_in, const int* in_sizes, int n_in,
                              void* d_out, int out_size, void* d_ws, size_t ws_size,
                              hipStream_t stream) {
    const float* x     = (const float*)d_in[0];
    const int*   src   = (const int*)  d_in[1];
    const int*   dst   = (const int*)  d_in[2];
    const float* W0    = (const float*)d_in[3];
    const float* al0   = (const float*)d_in[4];
    const float* ar0   = (const float*)d_in[5];
    const float* b0    = (const float*)d_in[6];
    const float* W1    = (const float*)d_in[7];
    const float* al1   = (const float*)d_in[8];
    const float* ar1   = (const float*)d_in[9];
    const float* b1    = (const float*)d_in[10];
    const float* noise = (const float*)d_in[11];
    float* out = (float*)d_out;

    const int IN_F = 128, HID = 64, HEADS = 4, OUT_F = 64;
    const int F0 = HEADS * HID;                  // 256
    const int N  = in_sizes[0] / IN_F;           // 100000
    const int E  = in_sizes[1];                  // 900000

    float* h0    = (float*)d_ws;                 // [N,256] layer-0 features (later reused as g1)
    float* acc0  = h0   + (size_t)N * F0;        // [N,256] layer-0 accumulator -> h1
    float* abuf  = acc0 + (size_t)N * F0;        // [E,4]  per-edge exp cache
    float* el    = abuf + (size_t)E * HEADS;     // [N,4]
    float* er    = el   + (size_t)N * HEADS;     // [N,4]
    float* emax  = er   + (size_t)N * HEADS;     // [N,4]
    float* denom = emax + (size_t)N * HEADS;     // [N,4]
    float* Bp0   = denom + (size_t)N * HEADS;    // [IN_F/2, F0] float2  (32K floats)
    float* Bp1   = Bp0 + (size_t)IN_F * F0;      // [F0/2, OUT_F] float2 (16K floats)

    auto gridFor = [](long n) { return (unsigned)((n + 255) / 256); };
    const float nscale = sqrtf(2.0f * logf(1.25f / 0.05f));

    // weight repacks (tiny, L2-resident)
    repack_B_kernel<<<gridFor((long)(IN_F / 2) * F0), 256, 0, stream>>>(W0, Bp0, IN_F, F0);
    repack_B_kernel<<<gridFor((long)(F0 / 2) * OUT_F), 256, 0, stream>>>(W1, Bp1, F0, OUT_F);

    // ---------------- layer 0 ----------------
    {   // h0 = x @ W0   (32x64 tile per wave, 8 waves per block)
        int tiles = (N / 32) * (F0 / 64);
        gemm_wmma_f32<<<(tiles + 7) / 8, 256, 0, stream>>>(x, Bp0, h0, N, IN_F, F0);
    }
    attn_scores_kernel<<<gridFor((long)N * HEADS), 256, 0, stream>>>(h0, al0, ar0, el, er, N, HEADS, HID);
    fill_kernel<<<gridFor((long)N * HEADS), 256, 0, stream>>>(emax, (long)N * HEADS, -INFINITY);
    fill_kernel<<<gridFor((long)N * HEADS), 256, 0, stream>>>(denom, (long)N * HEADS, 0.f);
    fill_kernel<<<gridFor((long)N * F0), 256, 0, stream>>>(acc0, (long)N * F0, 0.f);
    edge_max_kernel<<<gridFor((long)E * HEADS), 256, 0, stream>>>(src, dst, el, er, emax, (long)E * HEADS, HEADS);
    edge_exp_kernel<<<gridFor((long)E * HEADS), 256, 0, stream>>>(src, dst, el, er, emax, denom, abuf, (long)E * HEADS, HEADS);
    edge_scatter_kernel<<<gridFor((long)E * (F0 >> 2)), 256, 0, stream>>>(src, dst, abuf, denom, h0, acc0, E, HEADS, HID);
    h1_kernel<<<gridFor((long)N * F0), 256, 0, stream>>>(acc0, b0, noise, nscale, (long)N * F0, F0);

    // ---------------- layer 1 ----------------
    float* g1 = h0;                              // reuse h0 region: g1 = h1 @ W1  [N,64]
    {
        int tiles = (N / 32) * (OUT_F / 64);
        gemm_wmma_f32<<<(tiles + 7) / 8, 256, 0, stream>>>(acc0, Bp1, g1, N, F0, OUT_F);
    }
    attn_scores_kernel<<<gridFor((long)N), 256, 0, stream>>>(g1, al1, ar1, el, er, N, 1, OUT_F);
    fill_kernel<<<gridFor((long)N), 256, 0, stream>>>(emax, (long)N, -INFINITY);
    fill_kernel<<<gridFor((long)N), 256, 0, stream>>>(denom, (long)N, 0.f);
    bias_init_kernel<<<gridFor((long)N * OUT_F), 256, 0, stream>>>(out, b1, (long)N * OUT_F, OUT_F);
    edge_max_kernel<<<gridFor((long)E), 256, 0, stream>>>(src, dst, el, er, emax, (long)E, 1);
    edge_exp_kernel<<<gridFor((long)E), 256, 0, stream>>>(src, dst, el, er, emax, denom, abuf, (long)E, 1);
    edge_scatter_kernel<<<gridFor((long)E * (OUT_F >> 2)), 256, 0, stream>>>(src, dst, abuf, denom, g1, out, E, 1, OUT_F);
}